// VectorQuantizerEMA_67714454389128
// MI455X (gfx1250) — compile-verified
//
#include <hip/hip_runtime.h>
#include <hip/hip_bf16.h>
#include <math.h>

typedef __attribute__((ext_vector_type(2))) float v2f;
typedef __attribute__((ext_vector_type(8))) float v8f;

#define N_PTS   131072
#define K_CB    2048
#define D_DIM   64
#define DECAYF  0.99f
#define OMDECAY 0.01f
#define EPSF    1e-5f

// ---------------- workspace layout (floats) ----------------
#define WS_W2     0
#define WS_COUNTS 2048
#define WS_CS     4096
#define WS_LOSS   6144
#define WS_DW     6160
#define WS_IDX    (6160 + K_CB * D_DIM)

// ---------------- output layout (floats) ----------------
#define OUT_Q    0
#define OUT_PERP (N_PTS * D_DIM)
#define OUT_LOSS (OUT_PERP + 1)
#define OUT_CS   (OUT_LOSS + 1)
#define OUT_EMA  (OUT_CS + K_CB)
#define OUT_WN   (OUT_EMA + K_CB * D_DIM)

// ======== kernel 0: zero the scratch accumulators ========
__global__ void vq_zero_kernel(float* ws) {
    int i = blockIdx.x * blockDim.x + threadIdx.x;
    int n = WS_DW + K_CB * D_DIM - WS_COUNTS;
    if (i < n) ws[WS_COUNTS + i] = 0.0f;
}

// ======== kernel 1: w2[k] = sum_d w[k][d]^2 ========
__global__ void vq_w2_kernel(const float* __restrict__ w, float* __restrict__ ws) {
    int k = blockIdx.x * blockDim.x + threadIdx.x;
    if (k >= K_CB) return;
    const float4* row = (const float4*)(w + (size_t)k * D_DIM);
    float s = 0.0f;
#pragma unroll
    for (int j = 0; j < D_DIM / 4; ++j) {
        float4 v = row[j];
        s += v.x * v.x + v.y * v.y + v.z * v.z + v.w * v.w;
    }
    ws[WS_W2 + k] = s;
}

// ======== kernel 2: fused distance GEMM + row argmin (WMMA f32) ========
// block = 256 threads = 8 waves; each wave owns 32 rows (two 16-row A strips
// kept in VGPRs); block covers 256 rows. Every B fragment read from LDS feeds
// TWO independent WMMA accumulation chains (halves DS traffic per matrix op,
// doubles WMMA ILP). Weight staged through LDS in 64-column tiles with a
// 68-float padded column stride (bank-conflict-free ds_read_b64 B pattern).
#define TILE_COLS 64
#define LDS_STRIDE 68

__global__ __launch_bounds__(256) void vq_argmin_kernel(
    const float* __restrict__ x, const float* __restrict__ w,
    const float* __restrict__ ws_ro, int* __restrict__ idx_out) {
    __shared__ float lw[TILE_COLS * LDS_STRIDE];
    __shared__ float lw2[TILE_COLS];

    const int tid  = threadIdx.x;
    const int lane = tid & 31;
    const int wave = tid >> 5;
    const int ll   = lane & 15;   // column / row-within-half selector
    const int hl   = lane >> 4;   // half selector (K=0,1 vs K=2,3 for A/B)

    const int row0 = blockIdx.x * 256 + wave * 32;

    // ---- A operands: two 16-row strips of x, ISA 16x4 f32 A layout ----
    // lane ll (half hl) supplies row, elements d = 4c + 2*hl, +1
    v2f areg0[16], areg1[16];
    {
        const float* arow0 = x + (size_t)(row0 + ll) * D_DIM + 2 * hl;
        const float* arow1 = arow0 + 16 * D_DIM;
#pragma unroll
        for (int c = 0; c < 16; ++c) {
            areg0[c] = *(const v2f*)(arow0 + 4 * c);
            areg1[c] = *(const v2f*)(arow1 + 4 * c);
        }
    }

    float bestv0[8], bestv1[8];
    int   besti0[8], besti1[8];
#pragma unroll
    for (int i = 0; i < 8; ++i) {
        bestv0[i] = 3.4e38f; besti0[i] = 0;
        bestv1[i] = 3.4e38f; besti1[i] = 0;
    }

    const float* w2 = ws_ro + WS_W2;

    for (int t = 0; t < K_CB / TILE_COLS; ++t) {
        __syncthreads();  // previous tile fully consumed
        // cooperative load: 64 cols x 64 floats, 16 floats (4x float4) / thread
        {
            const int col = tid >> 2;
            const int seg = (tid & 3) * 16;
            const float4* src = (const float4*)(w + ((size_t)(t * TILE_COLS + col)) * D_DIM + seg);
            float4* dst = (float4*)(&lw[col * LDS_STRIDE + seg]);
#pragma unroll
            for (int j = 0; j < 4; ++j) dst[j] = src[j];
            if (tid < TILE_COLS) lw2[tid] = w2[t * TILE_COLS + tid];
        }
        // prefetch next weight tile toward L2 while we compute
        if (t + 1 < K_CB / TILE_COLS)
            __builtin_prefetch(w + ((size_t)(t + 1) * TILE_COLS) * D_DIM + tid * 16, 0, 1);
        __syncthreads();

#pragma unroll
        for (int s = 0; s < TILE_COLS / 16; ++s) {
            const int colb = t * TILE_COLS + s * 16;
            v8f acc0 = {0.0f, 0.0f, 0.0f, 0.0f, 0.0f, 0.0f, 0.0f, 0.0f};
            v8f acc1 = {0.0f, 0.0f, 0.0f, 0.0f, 0.0f, 0.0f, 0.0f, 0.0f};
            const float* bcol = &lw[(s * 16 + ll) * LDS_STRIDE + 2 * hl];
#pragma unroll
            for (int c = 0; c < 16; ++c) {
                v2f b = *(const v2f*)(bcol + 4 * c);
                acc0 = __builtin_amdgcn_wmma_f32_16x16x4_f32(
                    false, areg0[c], false, b, (short)0, acc0, false, false);
                acc1 = __builtin_amdgcn_wmma_f32_16x16x4_f32(
                    false, areg1[c], false, b, (short)0, acc1, false, false);
            }
            // dist(row, col) = w2[col] - 2*acc  (x^2 term constant per row)
            const float w2n = lw2[s * 16 + ll];
            const int   col = colb + ll;
#pragma unroll
            for (int i = 0; i < 8; ++i) {
                float dv0 = fmaf(-2.0f, acc0[i], w2n);
                if (dv0 < bestv0[i]) { bestv0[i] = dv0; besti0[i] = col; }
                float dv1 = fmaf(-2.0f, acc1[i], w2n);
                if (dv1 < bestv1[i]) { bestv1[i] = dv1; besti1[i] = col; }
            }
        }
    }

    // reduce min/argmin across the 16 lanes holding the same row
    // (masks < 16 keep the two halves independent under width-32 shuffles)
#pragma unroll
    for (int i = 0; i < 8; ++i) {
#pragma unroll
        for (int m = 1; m <= 8; m <<= 1) {
            float ov0 = __shfl_xor(bestv0[i], m, 32);
            int   oi0 = __shfl_xor(besti0[i], m, 32);
            if (ov0 < bestv0[i] || (ov0 == bestv0[i] && oi0 < besti0[i])) {
                bestv0[i] = ov0; besti0[i] = oi0;
            }
            float ov1 = __shfl_xor(bestv1[i], m, 32);
            int   oi1 = __shfl_xor(besti1[i], m, 32);
            if (ov1 < bestv1[i] || (ov1 == bestv1[i] && oi1 < besti1[i])) {
                bestv1[i] = ov1; besti1[i] = oi1;
            }
        }
    }
    if (ll == 0) {
#pragma unroll
        for (int i = 0; i < 8; ++i) {
            idx_out[row0 + 8 * hl + i]      = besti0[i];  // strip0: VGPR i <-> row i+8*hl
            idx_out[row0 + 16 + 8 * hl + i] = besti1[i];  // strip1
        }
    }
}

// ======== kernel 3: gather quantized, scatter counts/dw, loss partials ========
__global__ __launch_bounds__(256) void vq_scatter_kernel(
    const float* __restrict__ x, const float* __restrict__ w,
    const int* __restrict__ idx, float* __restrict__ qout, float* ws) {
    const int lane  = threadIdx.x & 31;
    const int gwave = blockIdx.x * 8 + (threadIdx.x >> 5);
    const int nwave = gridDim.x * 8;
    float* counts = ws + WS_COUNTS;
    float* dw     = ws + WS_DW;

    float lsum = 0.0f;
    for (int r = gwave; r < N_PTS; r += nwave) {
        const int k = idx[r];
        const size_t xo = (size_t)r * D_DIM;
        const size_t wo = (size_t)k * D_DIM;
        float x0 = x[xo + lane], x1 = x[xo + lane + 32];
        float q0 = w[wo + lane], q1 = w[wo + lane + 32];
        qout[xo + lane] = q0;
        qout[xo + lane + 32] = q1;
        float d0 = q0 - x0, d1 = q1 - x1;
        lsum += d0 * d0 + d1 * d1;
        atomicAdd(&dw[wo + lane], x0);
        atomicAdd(&dw[wo + lane + 32], x1);
        if (lane == 0) atomicAdd(&counts[k], 1.0f);
    }
#pragma unroll
    for (int m = 16; m >= 1; m >>= 1) lsum += __shfl_xor(lsum, m, 32);
    if (lane == 0) atomicAdd(ws + WS_LOSS, lsum);
}

// ======== kernel 4: cs normalization, perplexity, loss (single block) ========
__global__ void vq_stats_kernel(const float* __restrict__ ema_cs, float* ws,
                                float* __restrict__ out) {
    __shared__ float red[256];
    const int tid = threadIdx.x;
    const float* counts = ws + WS_COUNTS;

    float nloc = 0.0f;
    for (int k = tid; k < K_CB; k += 256)
        nloc += ema_cs[k] * DECAYF + OMDECAY * counts[k];
    red[tid] = nloc;
    __syncthreads();
    for (int s = 128; s >= 1; s >>= 1) {
        if (tid < s) red[tid] += red[tid + s];
        __syncthreads();
    }
    const float n = red[0];
    __syncthreads();

    for (int k = tid; k < K_CB; k += 256) {
        float csr = ema_cs[k] * DECAYF + OMDECAY * counts[k];
        float cs  = (csr + EPSF) / (n + K_CB * EPSF) * n;
        out[OUT_CS + k] = cs;
        ws[WS_CS + k]   = cs;
    }

    float sl = 0.0f;
    for (int k = tid; k < K_CB; k += 256) {
        float p = counts[k] * (1.0f / (float)N_PTS);
        sl += p * logf(p + 1e-10f);
    }
    red[tid] = sl;
    __syncthreads();
    for (int s = 128; s >= 1; s >>= 1) {
        if (tid < s) red[tid] += red[tid + s];
        __syncthreads();
    }
    if (tid == 0) {
        out[OUT_PERP] = expf(-red[0]);
        out[OUT_LOSS] = ws[WS_LOSS] / (float)((size_t)N_PTS * D_DIM);
    }
}

// ======== kernel 5: EMA weight update ========
__global__ void vq_ema_kernel(const float* __restrict__ ema_w, const float* ws,
                              float* __restrict__ out) {
    int i = blockIdx.x * blockDim.x + threadIdx.x;
    if (i >= K_CB * D_DIM) return;
    float e = ema_w[i] * DECAYF + OMDECAY * ws[WS_DW + i];
    out[OUT_EMA + i] = e;
    out[OUT_WN + i]  = e / ws[WS_CS + i / D_DIM];
}

extern "C" void kernel_launch(void* const* d_in, const int* in_sizes, int n_in,
                              void* d_out, int out_size, void* d_ws, size_t ws_size,
                              hipStream_t stream) {
    const float* x       = (const float*)d_in[0];  // [N, D]
    const float* w       = (const float*)d_in[1];  // [K, D]
    const float* ema_w   = (const float*)d_in[2];  // [K, D]
    const float* ema_cs  = (const float*)d_in[3];  // [K]
    float* out = (float*)d_out;
    float* ws  = (float*)d_ws;
    int* idx   = (int*)(ws + WS_IDX);

    {
        int n = WS_DW + K_CB * D_DIM - WS_COUNTS;
        vq_zero_kernel<<<(n + 255) / 256, 256, 0, stream>>>(ws);
    }
    vq_w2_kernel<<<K_CB / 256, 256, 0, stream>>>(w, ws);
    vq_argmin_kernel<<<N_PTS / 256, 256, 0, stream>>>(x, w, ws, idx);
    vq_scatter_kernel<<<512, 256, 0, stream>>>(x, w, idx, out + OUT_Q, ws);
    vq_stats_kernel<<<1, 256, 0, stream>>>(ema_cs, ws, out);
    vq_ema_kernel<<<(K_CB * D_DIM) / 256, 256, 0, stream>>>(ema_w, ws, out);
}